// FPLGNN_23038204575898
// MI455X (gfx1250) — compile-verified
//
#include <hip/hip_runtime.h>
#include <hip/hip_bf16.h>
#include <math.h>

// GAT 2-layer forward for gfx1250 (MI455X).
// Phase 1: h1 = x @ W1 via V_WMMA_F32_16X16X4_F32 (fp32 matrix path, W1 in LDS).
// Phase 2: per-node attention logits; 3-pass atomic edge softmax in L2-resident
//          scratch; channel-parallel scatter-add.
// Phase 3: fused ReLU + 64->1 projection + layer-2 logits; 1-head edge softmax;
//          scatter into d_out (init to b2).

typedef float v2f __attribute__((ext_vector_type(2)));
typedef float v8f __attribute__((ext_vector_type(8)));

#define NEG_SLOPE 0.2f

__device__ __forceinline__ float leaky(float v) {
    return v >= 0.0f ? v : NEG_SLOPE * v;
}

// Exact float atomic max via sign-split int/uint atomics.
// Requires destination initialized to -inf bit pattern (0xFF800000).
__device__ __forceinline__ void atomicMaxF(float* addr, float val) {
    if (val >= 0.0f) {
        atomicMax(reinterpret_cast<int*>(addr), __float_as_int(val));
    } else {
        atomicMin(reinterpret_cast<unsigned int*>(addr), __float_as_uint(val));
    }
}

// Edge (src,dst) with implicit self-loops appended after the E real edges.
__device__ __forceinline__ void edge_sd(const int* __restrict__ ei, long long E,
                                        long long e, int& s, int& d) {
    if (e < E) {
        s = ei[e];
        d = ei[E + e];
    } else {
        s = d = (int)(e - E);
    }
}

// ---------------------------------------------------------------------------
// k_init: zero accumulators, set segment-max buffers to -inf, seed d_out = b2.
// ---------------------------------------------------------------------------
__global__ void k_init(float* __restrict__ out1, float* __restrict__ mx1,
                       float* __restrict__ dn1, float* __restrict__ mx2,
                       float* __restrict__ dn2, float* __restrict__ out,
                       const float* __restrict__ b2, int N) {
    int i = blockIdx.x * blockDim.x + threadIdx.x;
    long long n64 = (long long)N * 64;
    if (i < n64) out1[i] = 0.0f;
    if (i < N * 4) {
        mx1[i] = __int_as_float(0xFF800000);  // -inf
        dn1[i] = 0.0f;
    }
    if (i < N) {
        mx2[i] = __int_as_float(0xFF800000);
        dn2[i] = 0.0f;
        out[i] = b2[0];
    }
}

// ---------------------------------------------------------------------------
// k_gemm1: h1[N,64] = x[N,128] @ W1[128,64] using fp32 WMMA 16x16x4.
// Block = 128 threads = 4 waves; block computes 16 rows x 64 cols.
// Wave w owns the 16x16 tile at columns [16w, 16w+16).
// A layout (16x4 f32): lanes 0-15 -> (row=lane, K=0..1), lanes 16-31 -> K=2..3.
// B layout (4x16 f32): VGPR0 = K row kb, VGPR1 = K row kb+1, N striped on lanes.
// D layout (16x16 f32): VGPR j -> row j (lanes 0-15) / row j+8 (lanes 16-31).
// ---------------------------------------------------------------------------
__global__ __launch_bounds__(128) void k_gemm1(const float* __restrict__ x,
                                               const float* __restrict__ W,
                                               float* __restrict__ h, int N) {
    __shared__ float wlds[128 * 64];  // 32 KB of 320 KB/WGP
    const int tid = threadIdx.x;
    for (int i = tid; i < 128 * 64; i += 128) wlds[i] = W[i];
    __syncthreads();

    const int wave = tid >> 5;
    const int lane = tid & 31;
    const int r0 = blockIdx.x * 16;
    const int c0 = wave * 16;
    const int rowcol = lane & 15;       // A row / B,D column index
    const int kb = (lane >> 4) << 1;    // 0 for lanes 0-15, 2 for lanes 16-31

    int rr = r0 + rowcol;
    if (rr >= N) rr = N - 1;            // clamp (N is a multiple of 16 anyway)
    const float* __restrict__ xr = x + (long long)rr * 128 + kb;
    const float* w0 = &wlds[kb * 64 + c0 + rowcol];

    v8f acc = {};
    for (int k = 0; k < 128; k += 4) {
        v2f a = *reinterpret_cast<const v2f*>(xr + k);  // 8B aligned (kb,k even)
        v2f b;
        b.x = w0[k * 64];
        b.y = w0[k * 64 + 64];
        acc = __builtin_amdgcn_wmma_f32_16x16x4_f32(
            /*neg_a=*/false, a, /*neg_b=*/false, b,
            /*c_mod=*/(short)0, acc, /*reuse_a=*/false, /*reuse_b=*/false);
    }

    const int hi = (lane >> 4) * 8;
    for (int j = 0; j < 8; ++j) {
        int r = r0 + hi + j;
        if (r < N) h[(long long)r * 64 + c0 + rowcol] = acc[j];
    }
}

// ---------------------------------------------------------------------------
// k_logits1: per-node attention logits for 4 heads x 16 channels.
// ---------------------------------------------------------------------------
__global__ void k_logits1(const float* __restrict__ h,
                          const float* __restrict__ asrc,
                          const float* __restrict__ adst,
                          float* __restrict__ ls, float* __restrict__ ld, int N) {
    int n = blockIdx.x * blockDim.x + threadIdx.x;
    if (n >= N) return;
    const float* __restrict__ hr = h + (long long)n * 64;
    #pragma unroll
    for (int hh = 0; hh < 4; ++hh) {
        float s = 0.0f, d = 0.0f;
        #pragma unroll
        for (int c = 0; c < 16; ++c) {
            float v = hr[hh * 16 + c];
            s += v * asrc[hh * 16 + c];
            d += v * adst[hh * 16 + c];
        }
        ls[n * 4 + hh] = s;
        ld[n * 4 + hh] = d;
    }
}

// ---------------------------------------------------------------------------
// Layer-1 edge softmax passes (4 heads per edge-thread).
// ---------------------------------------------------------------------------
__global__ void k_edge_max1(const int* __restrict__ ei, long long E, long long ET,
                            const float* __restrict__ ls,
                            const float* __restrict__ ld,
                            float* __restrict__ mx) {
    long long e = (long long)blockIdx.x * blockDim.x + threadIdx.x;
    if (e >= ET) return;
    int s, d;
    edge_sd(ei, E, e, s, d);
    #pragma unroll
    for (int hh = 0; hh < 4; ++hh) {
        float v = leaky(ls[(long long)s * 4 + hh] + ld[(long long)d * 4 + hh]);
        atomicMaxF(&mx[(long long)d * 4 + hh], v);
    }
}

__global__ void k_edge_sum1(const int* __restrict__ ei, long long E, long long ET,
                            const float* __restrict__ ls,
                            const float* __restrict__ ld,
                            const float* __restrict__ mx,
                            float* __restrict__ dn) {
    long long e = (long long)blockIdx.x * blockDim.x + threadIdx.x;
    if (e >= ET) return;
    int s, d;
    edge_sd(ei, E, e, s, d);
    #pragma unroll
    for (int hh = 0; hh < 4; ++hh) {
        float v = leaky(ls[(long long)s * 4 + hh] + ld[(long long)d * 4 + hh]);
        float ex = expf(v - mx[(long long)d * 4 + hh]);
        atomicAdd(&dn[(long long)d * 4 + hh], ex);
    }
}

// One thread per (edge, channel): 64 channels/edge, alpha recomputed per head.
__global__ void k_edge_scat1(const int* __restrict__ ei, long long E, long long ET,
                             const float* __restrict__ ls,
                             const float* __restrict__ ld,
                             const float* __restrict__ mx,
                             const float* __restrict__ dn,
                             const float* __restrict__ h,
                             float* __restrict__ out1) {
    long long t = (long long)blockIdx.x * blockDim.x + threadIdx.x;
    if (t >= ET * 64) return;
    long long e = t >> 6;
    int c = (int)(t & 63);
    int hh = c >> 4;
    int s, d;
    edge_sd(ei, E, e, s, d);
    float v = leaky(ls[(long long)s * 4 + hh] + ld[(long long)d * 4 + hh]);
    float alpha = expf(v - mx[(long long)d * 4 + hh]) /
                  (dn[(long long)d * 4 + hh] + 1e-16f);
    atomicAdd(&out1[(long long)d * 64 + c], h[(long long)s * 64 + c] * alpha);
}

// ---------------------------------------------------------------------------
// k_proj2: fused (out1 + b1) -> ReLU -> @W2[64,1] -> layer-2 logits.
// ---------------------------------------------------------------------------
__global__ void k_proj2(const float* __restrict__ out1,
                        const float* __restrict__ b1,
                        const float* __restrict__ W2,
                        const float* __restrict__ as2,
                        const float* __restrict__ ad2,
                        float* __restrict__ hp, float* __restrict__ ls2,
                        float* __restrict__ ld2, int N) {
    int n = blockIdx.x * blockDim.x + threadIdx.x;
    if (n >= N) return;
    const float* __restrict__ row = out1 + (long long)n * 64;
    float acc = 0.0f;
    #pragma unroll
    for (int c = 0; c < 64; ++c) {
        float v = row[c] + b1[c];
        v = v > 0.0f ? v : 0.0f;  // ReLU
        acc += v * W2[c];
    }
    hp[n] = acc;
    ls2[n] = acc * as2[0];
    ld2[n] = acc * ad2[0];
}

// ---------------------------------------------------------------------------
// Layer-2 edge softmax passes (single head).
// ---------------------------------------------------------------------------
__global__ void k_edge_max2(const int* __restrict__ ei, long long E, long long ET,
                            const float* __restrict__ ls,
                            const float* __restrict__ ld,
                            float* __restrict__ mx) {
    long long e = (long long)blockIdx.x * blockDim.x + threadIdx.x;
    if (e >= ET) return;
    int s, d;
    edge_sd(ei, E, e, s, d);
    atomicMaxF(&mx[d], leaky(ls[s] + ld[d]));
}

__global__ void k_edge_sum2(const int* __restrict__ ei, long long E, long long ET,
                            const float* __restrict__ ls,
                            const float* __restrict__ ld,
                            const float* __restrict__ mx,
                            float* __restrict__ dn) {
    long long e = (long long)blockIdx.x * blockDim.x + threadIdx.x;
    if (e >= ET) return;
    int s, d;
    edge_sd(ei, E, e, s, d);
    atomicAdd(&dn[d], expf(leaky(ls[s] + ld[d]) - mx[d]));
}

__global__ void k_edge_scat2(const int* __restrict__ ei, long long E, long long ET,
                             const float* __restrict__ ls,
                             const float* __restrict__ ld,
                             const float* __restrict__ mx,
                             const float* __restrict__ dn,
                             const float* __restrict__ hp,
                             float* __restrict__ out) {
    long long e = (long long)blockIdx.x * blockDim.x + threadIdx.x;
    if (e >= ET) return;
    int s, d;
    edge_sd(ei, E, e, s, d);
    float alpha = expf(leaky(ls[s] + ld[d]) - mx[d]) / (dn[d] + 1e-16f);
    atomicAdd(&out[d], hp[s] * alpha);
}

// ---------------------------------------------------------------------------
extern "C" void kernel_launch(void* const* d_in, const int* in_sizes, int n_in,
                              void* d_out, int out_size, void* d_ws, size_t ws_size,
                              hipStream_t stream) {
    const float* x   = (const float*)d_in[0];
    const int*   ei  = (const int*)d_in[1];
    const float* W1  = (const float*)d_in[2];
    const float* as1 = (const float*)d_in[3];
    const float* ad1 = (const float*)d_in[4];
    const float* b1  = (const float*)d_in[5];
    const float* W2  = (const float*)d_in[6];
    const float* as2 = (const float*)d_in[7];
    const float* ad2 = (const float*)d_in[8];
    const float* b2  = (const float*)d_in[9];

    const int N = in_sizes[0] / 128;
    const long long E = (long long)in_sizes[1] / 2;
    const long long ET = E + N;  // with self-loops

    // Workspace layout (floats). Total = 149*N floats (~60 MB), L2-resident.
    float* ws   = (float*)d_ws;
    float* h1   = ws;                         // N*64
    float* out1 = h1 + (size_t)N * 64;        // N*64
    float* ls1  = out1 + (size_t)N * 64;      // N*4
    float* ld1  = ls1 + (size_t)N * 4;        // N*4
    float* mx1  = ld1 + (size_t)N * 4;        // N*4
    float* dn1  = mx1 + (size_t)N * 4;        // N*4
    float* hp   = dn1 + (size_t)N * 4;        // N
    float* ls2  = hp + N;                     // N
    float* ld2  = ls2 + N;                    // N
    float* mx2  = ld2 + N;                    // N
    float* dn2  = mx2 + N;                    // N
    float* out  = (float*)d_out;              // N

    const int T = 256;
    const long long n64 = (long long)N * 64;

    k_init<<<(unsigned)((n64 + T - 1) / T), T, 0, stream>>>(
        out1, mx1, dn1, mx2, dn2, out, b2, N);

    k_gemm1<<<(unsigned)((N + 15) / 16), 128, 0, stream>>>(x, W1, h1, N);

    k_logits1<<<(unsigned)((N + T - 1) / T), T, 0, stream>>>(
        h1, as1, ad1, ls1, ld1, N);

    unsigned gE = (unsigned)((ET + T - 1) / T);
    k_edge_max1<<<gE, T, 0, stream>>>(ei, E, ET, ls1, ld1, mx1);
    k_edge_sum1<<<gE, T, 0, stream>>>(ei, E, ET, ls1, ld1, mx1, dn1);

    unsigned gEC = (unsigned)((ET * 64 + T - 1) / T);
    k_edge_scat1<<<gEC, T, 0, stream>>>(ei, E, ET, ls1, ld1, mx1, dn1, h1, out1);

    k_proj2<<<(unsigned)((N + T - 1) / T), T, 0, stream>>>(
        out1, b1, W2, as2, ad2, hp, ls2, ld2, N);

    k_edge_max2<<<gE, T, 0, stream>>>(ei, E, ET, ls2, ld2, mx2);
    k_edge_sum2<<<gE, T, 0, stream>>>(ei, E, ET, ls2, ld2, mx2, dn2);
    k_edge_scat2<<<gE, T, 0, stream>>>(ei, E, ET, ls2, ld2, mx2, dn2, hp, out);
}